// DCLoss_56487409877002
// MI455X (gfx1250) — compile-verified
//
#include <hip/hip_runtime.h>

// Dark-channel prior loss on MI455X (gfx1250).
// dc = maxpool_35x35(max_c(1-img)) == 1 - minpool_35x35(min_c(img)); out = mean|dc|.
// HBM traffic is at the ~100.7MB compulsory floor (image is L2-resident; halo
// re-reads hit the 192MB L2), so the LDS pooling phases are the next bottleneck:
// all in-tile phases use 16B-aligned float4 (ds b128) accesses on padded strides.
// CDNA5 paths: global_load_async_to_lds_b32 (ASYNCcnt), s_wait_asynccnt,
// global_prefetch_b8, 320KB/WGP LDS (124KB static per WG), wave32 sizing.

#define N_IMG 32
#define H 512
#define W 512
#define HW (H * W)
#define PADR 17
#define OTY 128                   // output tile rows per block
#define OTX 64                    // output tile cols per block
#define TR (OTY + 2 * PADR)       // 162 input rows in LDS tile
#define TC (OTX + 2 * PADR)       // 98 input cols in LDS tile (logical)
#define SAP 100                   // padded sA row stride (16B-aligned float4 rows)
#define M5C (OTX + 30)            // 94 horizontal min-of-5 runs per row (logical)
#define SBP 96                    // padded sB row stride
#define M5R (OTY + 30)            // 158 vertical min-of-5 runs per col
#define TILES_X (W / OTX)         // 8
#define TILES_Y (H / OTY)         // 4
#define NPART (TILES_X * TILES_Y * N_IMG)  // 1024 block partials
#define NT 256

// Low 32 bits of a flat pointer into __shared__ == LDS byte offset (AS3 offset).
__device__ __forceinline__ unsigned lds_off_of(const void* p) {
  return (unsigned)(unsigned long long)p;
}

// Async DMA one dword global -> LDS (per-lane addresses), tracked by ASYNCcnt.
__device__ __forceinline__ void async_copy_b32(unsigned lds_off, const float* g) {
  asm volatile("global_load_async_to_lds_b32 %0, %1, off"
               :: "v"(lds_off), "v"(g)
               : "memory");
}

__device__ __forceinline__ void wait_async0() {
  asm volatile("s_wait_asynccnt 0" ::: "memory");
}

__device__ __forceinline__ float4 fmin4(float4 a, float4 b) {
  return make_float4(fminf(a.x, b.x), fminf(a.y, b.y),
                     fminf(a.z, b.z), fminf(a.w, b.w));
}

__launch_bounds__(NT)
__global__ void dc_tile_kernel(const float* __restrict__ img,
                               float* __restrict__ partial) {
  // 127 KB static LDS: 2 workgroups co-resident per 320KB WGP.
  __shared__ float sA[TR * SAP];  // channel-min tile; reused as hm[TR][OTX]
  __shared__ float sB[TR * SBP];  // min-of-5 runs;   reused as m5y[M5R][OTX]

  const int tid = threadIdx.x;
  const int n = blockIdx.y;
  const int ty = blockIdx.x / TILES_X;
  const int tx = blockIdx.x - ty * TILES_X;
  const int y0 = ty * OTY, x0 = tx * OTX;

  const float* base = img + (size_t)n * 3 * HW;  // channel 0 plane

  // ---- Phase 0a: async DMA channel-0 tile into LDS; +inf halo for OOB.
  // Prefetch c1/c2 lines so phase 0b hits cache while the DMA drains.
  for (int idx = tid; idx < TR * TC; idx += NT) {
    int r = idx / TC, c = idx - r * TC;
    int gy = y0 + r - PADR, gx = x0 + c - PADR;
    if ((unsigned)gy < (unsigned)H && (unsigned)gx < (unsigned)W) {
      const float* g0 = base + gy * W + gx;
      async_copy_b32(lds_off_of(&sA[r * SAP + c]), g0);
      __builtin_prefetch(g0 + HW, 0, 3);       // channel 1
      __builtin_prefetch(g0 + 2 * HW, 0, 3);   // channel 2
    } else {
      sA[r * SAP + c] = __builtin_inff();
    }
  }
  wait_async0();
  __syncthreads();

  // ---- Phase 0b: fold channels 1 and 2 -> sA = min_c(img) over the tile.
  for (int idx = tid; idx < TR * TC; idx += NT) {
    int r = idx / TC, c = idx - r * TC;
    int gy = y0 + r - PADR, gx = x0 + c - PADR;
    if ((unsigned)gy < (unsigned)H && (unsigned)gx < (unsigned)W) {
      const float* g = base + gy * W + gx;
      float v = fminf(g[HW], g[2 * HW]);
      sA[r * SAP + c] = fminf(sA[r * SAP + c], v);
    }
  }
  __syncthreads();

  // ---- Phase 1: horizontal min-of-5 runs (35 = 5*7). 4 outputs per unit:
  // 2 aligned b128 loads + 1 b128 store, pairwise-min reuse (11 v_min / 4 out).
  for (int g = tid; g < TR * 24; g += NT) {          // 24 float4 groups per row
    int r = g / 24, i0 = (g - r * 24) * 4;
    const float4* A4 = reinterpret_cast<const float4*>(&sA[r * SAP + i0]);
    float4 a0 = A4[0], a1 = A4[1];
    float c1 = fminf(a0.y, a0.z);
    float c2 = fminf(a0.w, a1.x);
    float c3 = fminf(a1.y, a1.z);
    float4 o;
    o.x = fminf(fminf(a0.x, c1), c2);
    o.y = fminf(fminf(c1, c2), a1.y);
    o.z = fminf(fminf(a0.z, c2), c3);
    o.w = fminf(fminf(c2, c3), a1.w);
    *reinterpret_cast<float4*>(&sB[r * SBP + i0]) = o;
  }
  __syncthreads();

  // ---- Phase 2: 35-wide row min = min of 7 runs (taps stride 5, unaligned):
  // load a contiguous 40-float register window per 8 outputs (10 b128 loads).
  for (int g = tid; g < TR * 8; g += NT) {           // 8 groups of 8 cols per row
    int r = g / 8, c0 = (g - r * 8) * 8;
    const float4* B4 = reinterpret_cast<const float4*>(&sB[r * SBP + c0]);
    float w[40];
#pragma unroll
    for (int t = 0; t < 10; ++t) {
      float4 q = B4[t];
      w[4 * t] = q.x; w[4 * t + 1] = q.y; w[4 * t + 2] = q.z; w[4 * t + 3] = q.w;
    }
    float o[8];
#pragma unroll
    for (int k = 0; k < 8; ++k) {
      float m = w[k];
#pragma unroll
      for (int j = 1; j < 7; ++j) m = fminf(m, w[k + 5 * j]);
      o[k] = m;
    }
    float* hm = &sA[r * OTX + c0];
    *reinterpret_cast<float4*>(hm)     = make_float4(o[0], o[1], o[2], o[3]);
    *reinterpret_cast<float4*>(hm + 4) = make_float4(o[4], o[5], o[6], o[7]);
  }
  __syncthreads();

  // ---- Phase 3: vertical min-of-5 runs -> sB[i][c], float4 column groups.
  for (int g = tid; g < M5R * 16; g += NT) {         // 16 float4 groups per row
    int i = g / 16, c0 = (g - i * 16) * 4;
    const float4* A4 = reinterpret_cast<const float4*>(&sA[i * OTX + c0]);
    float4 m = fmin4(fmin4(A4[0], A4[16]), fmin4(A4[32], A4[48]));
    m = fmin4(m, A4[64]);                            // row step = OTX/4 = 16
    *reinterpret_cast<float4*>(&sB[i * OTX + c0]) = m;
  }
  __syncthreads();

  // ---- Phase 4: 35-tall column min = min of 7 runs; accumulate |1 - m|.
  float acc = 0.0f;
  for (int g = tid; g < OTY * 16; g += NT) {
    int o = g / 16, c0 = (g - o * 16) * 4;
    const float4* B4 = reinterpret_cast<const float4*>(&sB[o * OTX + c0]);
    float4 m = B4[0];
#pragma unroll
    for (int j = 1; j < 7; ++j) m = fmin4(m, B4[j * 80]);  // 5 rows * 16 f4
    acc += fabsf(1.0f - m.x);
    acc += fabsf(1.0f - m.y);
    acc += fabsf(1.0f - m.z);
    acc += fabsf(1.0f - m.w);
  }
  __syncthreads();

  // ---- Phase 5: deterministic block tree reduction.
  sA[tid] = acc;
  __syncthreads();
  for (int s = NT / 2; s > 0; s >>= 1) {
    if (tid < s) sA[tid] += sA[tid + s];
    __syncthreads();
  }
  if (tid == 0) partial[blockIdx.y * gridDim.x + blockIdx.x] = sA[0];
}

__launch_bounds__(NT)
__global__ void dc_finalize_kernel(const float* __restrict__ partial,
                                   float* __restrict__ out) {
  __shared__ float s[NT];
  float acc = 0.0f;
  for (int i = threadIdx.x; i < NPART; i += NT) acc += partial[i];  // fixed order
  s[threadIdx.x] = acc;
  __syncthreads();
  for (int k = NT / 2; k > 0; k >>= 1) {
    if (threadIdx.x < k) s[threadIdx.x] += s[threadIdx.x + k];
    __syncthreads();
  }
  if (threadIdx.x == 0)
    out[0] = s[0] * (1.0f / (float)(N_IMG * H * W));  // mean over 8388608
}

extern "C" void kernel_launch(void* const* d_in, const int* in_sizes, int n_in,
                              void* d_out, int out_size, void* d_ws, size_t ws_size,
                              hipStream_t stream) {
  const float* img = (const float*)d_in[0];
  float* out = (float*)d_out;
  float* partial = (float*)d_ws;  // NPART * 4 = 4 KB scratch

  dim3 grid(TILES_X * TILES_Y, N_IMG);  // (32, 32) blocks
  dc_tile_kernel<<<grid, dim3(NT), 0, stream>>>(img, partial);
  dc_finalize_kernel<<<dim3(1), dim3(NT), 0, stream>>>(partial, out);
}